// GNN_node_40450001994247
// MI455X (gfx1250) — compile-verified
//
#include <hip/hip_runtime.h>
#include <hip/hip_bf16.h>

// ---------------------------------------------------------------------------
// GINE GNN for MI455X (gfx1250, wave32).  f32 end-to-end, GEMMs on
// v_wmma_f32_16x16x4_f32 with pair-packed, bank-swizzled LDS weight panels;
// edge scatter via relaxed agent-scope global_atomic_add_f32 with
// wave-uniform (scalar) index loads.
// ---------------------------------------------------------------------------

typedef __attribute__((ext_vector_type(2))) float v2f;
typedef __attribute__((ext_vector_type(8))) float v8f;

#define NNODES 50000
#define NEDGES 500000
#define DIM    128
#define NLAYER 5
#define GEMM_WAVES 5           // waves per GEMM block; 5 * 625 = 3125 m-tiles

static __device__ __forceinline__ v8f wmma_f32_16x16x4(v2f a, v2f b, v8f c) {
  // 8 args: (neg_a, A, neg_b, B, c_mod, C, reuse_a, reuse_b)
  return __builtin_amdgcn_wmma_f32_16x16x4_f32(false, a, false, b, (short)0, c,
                                               false, false);
}

// ---------------------------------------------------------------------------
// GEMM: C[M,Nc] = f(A)[M,K] @ B[K,Nc] + bias.
// Block = 5 waves; each wave computes a 16x64 C-tile (4 v8f accumulators).
// B panel staged into LDS pair-packed: element (k,c) lives in float2 slot
// [kp*64 + (c ^ ((kp&1)<<4))], kp = k/2, so a lane's B fragment
// (B[ka][c], B[ka+1][c]) is a single contiguous ds_load_b64.  The XOR-16
// column swizzle keyed on kp parity puts lanes 0-15 (even kp) and lanes
// 16-31 (odd kp) on disjoint LDS bank halves -> conflict-free.
// A-operand loads stay in global (L2-resident) with the elementwise
// prologue fused:
//   MODE 0: f(A) = A                      (encoder)
//   MODE 1: f(A) = A + A2                 (z = h + agg)
//   MODE 2: f(A) = relu(A*scale + shift)  (BN + ReLU fused into GEMM2)
// Fragment layouts per CDNA5 ISA 7.12.2:
//   A 16x4 f32 : lanes 0-15 -> K = k0+{0,1};  lanes 16-31 -> K = k0+{2,3}
//   B  4x16    : mirrored (VGPR0 row k0+2*half, VGPR1 row k0+2*half+1)
//   C 16x16    : VGPR i: lanes 0-15 -> (M=i, N=lane); lanes 16-31 -> (M=i+8)
// ---------------------------------------------------------------------------
template <int MODE>
__global__ __launch_bounds__(32 * GEMM_WAVES) void gemm_wmma_kernel(
    const float* __restrict__ A, const float* __restrict__ A2,
    const float* __restrict__ scale, const float* __restrict__ shift,
    const float* __restrict__ B, const float* __restrict__ bias,
    float* __restrict__ C, int M, int K, int Nc) {
  extern __shared__ float ldsB[];  // K*64 floats, pair-packed float2 slots

  const int tid  = threadIdx.x;
  const int wave = tid >> 5;      // 0..GEMM_WAVES-1
  const int lane = tid & 31;
  const int half = lane >> 4;     // 0 or 1
  const int l15  = lane & 15;
  const int n0 = blockIdx.y * 64;
  const int m0 = (blockIdx.x * GEMM_WAVES + wave) * 16;
  const int row = m0 + l15;       // A row this lane feeds

  // ---- stage B panel pair-packed + swizzled into LDS ----
  float2* __restrict__ ldsB2 = (float2*)ldsB;
  for (int i = tid; i < (K / 2) * 64; i += 32 * GEMM_WAVES) {
    const int kp = i >> 6;             // K row-pair
    const int c  = i & 63;             // panel column
    const int cs = c ^ ((kp & 1) << 4);
    float2 v;
    v.x = B[(size_t)(2 * kp)     * Nc + n0 + c];
    v.y = B[(size_t)(2 * kp + 1) * Nc + n0 + c];
    ldsB2[kp * 64 + cs] = v;
  }
  __syncthreads();

  v8f acc[4];
#pragma unroll
  for (int t = 0; t < 4; ++t) {
    const float bv = bias[n0 + t * 16 + l15];
#pragma unroll
    for (int i = 0; i < 8; ++i) acc[t][i] = bv;
  }

  const float* __restrict__ Arow = A + (size_t)row * K;

#pragma unroll 4
  for (int k0 = 0; k0 < K; k0 += 4) {
    const int ka = k0 + half * 2;
    v2f a;
    if (MODE == 1) {
      const float* __restrict__ A2row = A2 + (size_t)row * K;
      a[0] = Arow[ka]     + A2row[ka];
      a[1] = Arow[ka + 1] + A2row[ka + 1];
    } else if (MODE == 2) {
      a[0] = fmaxf(Arow[ka]     * scale[ka]     + shift[ka],     0.0f);
      a[1] = fmaxf(Arow[ka + 1] * scale[ka + 1] + shift[ka + 1], 0.0f);
    } else {
      a[0] = Arow[ka];
      a[1] = Arow[ka + 1];
    }
    const int kp = ka >> 1;                       // even for lanes 0-15, odd 16-31
    const float2* __restrict__ Bp = ldsB2 + kp * 64;
    const int sw = (kp & 1) << 4;
#pragma unroll
    for (int t = 0; t < 4; ++t) {
      const float2 bv = Bp[(t * 16 + l15) ^ sw];  // one ds_load_b64, no shuffle
      v2f b;
      b[0] = bv.x;
      b[1] = bv.y;
      acc[t] = wmma_f32_16x16x4(a, b, acc[t]);
    }
  }

#pragma unroll
  for (int t = 0; t < 4; ++t) {
    const int col = n0 + t * 16 + l15;
#pragma unroll
    for (int i = 0; i < 8; ++i) {
      const int r = m0 + i + half * 8;
      C[(size_t)r * Nc + col] = acc[t][i];
    }
  }
}

// ---------------------------------------------------------------------------
// Edge message + scatter: one wave per edge, lane = 4-float feature group.
// src/dst indices are wave-uniform (readfirstlane) -> scalar loads; bases
// become SGPR + lane offset.  m = relu(h[src]+edge_attr); agg[dst] += m via
// relaxed agent-scope f32 atomics (no return -> STOREcnt path).
// ---------------------------------------------------------------------------
__global__ __launch_bounds__(256) void edge_message_kernel(
    const float* __restrict__ h, const float* __restrict__ edge_attr,
    const int* __restrict__ src, const int* __restrict__ dst,
    float* __restrict__ agg, int E) {
  const int wv = blockIdx.x * (256 / 32) + (threadIdx.x >> 5);
  if (wv >= E) return;
  const int e = __builtin_amdgcn_readfirstlane(wv);
  const int lane = threadIdx.x & 31;   // feature group (DIM/4 == 32)
  const int s = src[e];                // uniform address -> s_load
  const int d = dst[e];

  const float4 hv = ((const float4*)(h + (size_t)s * DIM))[lane];
  const float4 ev = ((const float4*)(edge_attr + (size_t)e * DIM))[lane];
  float4 m;
  m.x = fmaxf(hv.x + ev.x, 0.0f);
  m.y = fmaxf(hv.y + ev.y, 0.0f);
  m.z = fmaxf(hv.z + ev.z, 0.0f);
  m.w = fmaxf(hv.w + ev.w, 0.0f);

  float* ap = agg + (size_t)d * DIM + lane * 4;
  __hip_atomic_fetch_add(ap + 0, m.x, __ATOMIC_RELAXED, __HIP_MEMORY_SCOPE_AGENT);
  __hip_atomic_fetch_add(ap + 1, m.y, __ATOMIC_RELAXED, __HIP_MEMORY_SCOPE_AGENT);
  __hip_atomic_fetch_add(ap + 2, m.z, __ATOMIC_RELAXED, __HIP_MEMORY_SCOPE_AGENT);
  __hip_atomic_fetch_add(ap + 3, m.w, __ATOMIC_RELAXED, __HIP_MEMORY_SCOPE_AGENT);
}

// ---------------------------------------------------------------------------
// BN column statistics, float4-vectorized.  256 threads cover C/4 column
// groups x (256/(C/4)) row sublanes; block-local accumulation then one
// atomic add per column per sublane.
// ---------------------------------------------------------------------------
__global__ __launch_bounds__(256) void bn_stats_kernel(
    const float* __restrict__ Z, int Nrows, int C, int rows_per_block,
    float* __restrict__ sum, float* __restrict__ sqsum) {
  const int cpg = C >> 2;                 // column groups (32 or 64)
  const int c4 = threadIdx.x & (cpg - 1); // column group id
  const int rsub = threadIdx.x / cpg;     // row sublane
  const int rstep = 256 / cpg;            // rows per pass (8 or 4)
  int r0 = blockIdx.x * rows_per_block;
  int r1 = r0 + rows_per_block;
  if (r1 > Nrows) r1 = Nrows;

  const float4* __restrict__ Z4 = (const float4*)Z;
  float4 s = make_float4(0.f, 0.f, 0.f, 0.f);
  float4 s2 = make_float4(0.f, 0.f, 0.f, 0.f);
  for (int r = r0 + rsub; r < r1; r += rstep) {
    const float4 v = Z4[(size_t)r * cpg + c4];
    s.x += v.x; s.y += v.y; s.z += v.z; s.w += v.w;
    s2.x += v.x * v.x; s2.y += v.y * v.y; s2.z += v.z * v.z; s2.w += v.w * v.w;
  }
  const int c = c4 * 4;
  __hip_atomic_fetch_add(&sum[c + 0], s.x, __ATOMIC_RELAXED, __HIP_MEMORY_SCOPE_AGENT);
  __hip_atomic_fetch_add(&sum[c + 1], s.y, __ATOMIC_RELAXED, __HIP_MEMORY_SCOPE_AGENT);
  __hip_atomic_fetch_add(&sum[c + 2], s.z, __ATOMIC_RELAXED, __HIP_MEMORY_SCOPE_AGENT);
  __hip_atomic_fetch_add(&sum[c + 3], s.w, __ATOMIC_RELAXED, __HIP_MEMORY_SCOPE_AGENT);
  __hip_atomic_fetch_add(&sqsum[c + 0], s2.x, __ATOMIC_RELAXED, __HIP_MEMORY_SCOPE_AGENT);
  __hip_atomic_fetch_add(&sqsum[c + 1], s2.y, __ATOMIC_RELAXED, __HIP_MEMORY_SCOPE_AGENT);
  __hip_atomic_fetch_add(&sqsum[c + 2], s2.z, __ATOMIC_RELAXED, __HIP_MEMORY_SCOPE_AGENT);
  __hip_atomic_fetch_add(&sqsum[c + 3], s2.w, __ATOMIC_RELAXED, __HIP_MEMORY_SCOPE_AGENT);
}

// scale = g * rsqrt(var + eps); shift = b - mu*scale   (biased variance)
__global__ void bn_finalize_kernel(const float* __restrict__ sum,
                                   const float* __restrict__ sqsum,
                                   const float* __restrict__ g,
                                   const float* __restrict__ b,
                                   float* __restrict__ scale,
                                   float* __restrict__ shift, int Nrows, int C) {
  const int c = blockIdx.x * blockDim.x + threadIdx.x;
  if (c >= C) return;
  const float inv_n = 1.0f / (float)Nrows;
  const float mu = sum[c] * inv_n;
  const float var = sqsum[c] * inv_n - mu * mu;
  const float sc = g[c] * rsqrtf(var + 1e-5f);
  scale[c] = sc;
  shift[c] = b[c] - mu * sc;
}

// h_out = h_in + maybe_relu(z2*scale + shift)
__global__ __launch_bounds__(256) void residual_bn_kernel(
    const float* __restrict__ z2, const float* __restrict__ scale,
    const float* __restrict__ shift, const float* __restrict__ h_in,
    float* __restrict__ h_out, int do_relu, long long total) {
  const long long i = (long long)blockIdx.x * blockDim.x + threadIdx.x;
  if (i >= total) return;
  const int c = (int)(i & (DIM - 1));
  float v = z2[i] * scale[c] + shift[c];
  if (do_relu) v = fmaxf(v, 0.0f);
  h_out[i] = h_in[i] + v;
}

__global__ __launch_bounds__(256) void zero_kernel(float* __restrict__ p,
                                                   long long n) {
  const long long i = (long long)blockIdx.x * blockDim.x + threadIdx.x;
  if (i < n) p[i] = 0.0f;
}

// ---------------------------------------------------------------------------
extern "C" void kernel_launch(void* const* d_in, const int* in_sizes, int n_in,
                              void* d_out, int out_size, void* d_ws,
                              size_t ws_size, hipStream_t stream) {
  const float* x         = (const float*)d_in[0];   // [N,128]
  const float* edge_attr = (const float*)d_in[1];   // [E,128]
  const int*   src       = (const int*)d_in[2];     // [E]
  const int*   dst       = (const int*)d_in[3];     // [E]
  const float* enc_W     = (const float*)d_in[4];   // [128,128]
  const float* enc_b     = (const float*)d_in[5];   // [128]
  const float* W1        = (const float*)d_in[6];   // [L,128,256]
  const float* b1        = (const float*)d_in[7];   // [L,256]
  const float* bn1_g     = (const float*)d_in[8];   // [L,256]
  const float* bn1_b     = (const float*)d_in[9];   // [L,256]
  const float* W2        = (const float*)d_in[10];  // [L,256,128]
  const float* b2        = (const float*)d_in[11];  // [L,128]
  const float* bn_g      = (const float*)d_in[12];  // [L,128]
  const float* bn_b      = (const float*)d_in[13];  // [L,128]

  const int N = NNODES, E = NEDGES;
  const long long nd = (long long)N * DIM;  // 6.4M floats

  // Workspace layout (floats)
  float* h     = (float*)d_ws;       // [N,128]
  float* agg   = h + nd;             // [N,128]
  float* z1    = agg + nd;           // [N,256]
  float* z2    = z1 + (long long)N * 2 * DIM;  // [N,128]
  float* sum   = z2 + nd;            // [256]
  float* sqsum = sum + 256;          // [256]
  float* scale = sqsum + 256;        // [256]
  float* shift = scale + 256;        // [256]

  const dim3 gemm_block(32 * GEMM_WAVES, 1, 1);
  const int gemm_gx = N / (16 * GEMM_WAVES);  // 625 exact
  const int rows_per_block = 256;
  const int stat_blocks = (N + rows_per_block - 1) / rows_per_block;

  // ---- encoder: h = x @ enc_W + enc_b ----
  gemm_wmma_kernel<0><<<dim3(gemm_gx, DIM / 64), gemm_block,
                        DIM * 64 * sizeof(float), stream>>>(
      x, nullptr, nullptr, nullptr, enc_W, enc_b, h, N, DIM, DIM);

  for (int l = 0; l < NLAYER; ++l) {
    // agg = 0
    zero_kernel<<<(int)((nd + 255) / 256), 256, 0, stream>>>(agg, nd);
    // m = relu(h[src] + edge_attr); agg[dst] += m
    edge_message_kernel<<<(E + 7) / 8, 256, 0, stream>>>(h, edge_attr, src,
                                                         dst, agg, E);
    // z1 = (h + agg) @ W1[l] + b1[l]     [N,256]
    gemm_wmma_kernel<1><<<dim3(gemm_gx, (2 * DIM) / 64), gemm_block,
                          DIM * 64 * sizeof(float), stream>>>(
        h, agg, nullptr, nullptr, W1 + (size_t)l * DIM * 2 * DIM,
        b1 + (size_t)l * 2 * DIM, z1, N, DIM, 2 * DIM);
    // BN1 stats over z1 columns (C = 256)
    zero_kernel<<<2, 256, 0, stream>>>(sum, 512);
    bn_stats_kernel<<<stat_blocks, 256, 0, stream>>>(z1, N, 2 * DIM,
                                                     rows_per_block, sum,
                                                     sqsum);
    bn_finalize_kernel<<<1, 2 * DIM, 0, stream>>>(
        sum, sqsum, bn1_g + (size_t)l * 2 * DIM, bn1_b + (size_t)l * 2 * DIM,
        scale, shift, N, 2 * DIM);
    // z2 = relu(bn1(z1)) @ W2[l] + b2[l]   [N,128]  (BN+ReLU fused into A)
    gemm_wmma_kernel<2><<<dim3(gemm_gx, DIM / 64), gemm_block,
                          2 * DIM * 64 * sizeof(float), stream>>>(
        z1, nullptr, scale, shift, W2 + (size_t)l * 2 * DIM * DIM,
        b2 + (size_t)l * DIM, z2, N, 2 * DIM, DIM);
    // BN2 stats over z2 columns (C = 128)
    zero_kernel<<<2, 256, 0, stream>>>(sum, 512);
    bn_stats_kernel<<<stat_blocks, 256, 0, stream>>>(z2, N, DIM,
                                                     rows_per_block, sum,
                                                     sqsum);
    bn_finalize_kernel<<<1, DIM, 0, stream>>>(sum, sqsum, bn_g + (size_t)l * DIM,
                                              bn_b + (size_t)l * DIM, scale,
                                              shift, N, DIM);
    // h = h + maybe_relu(bn2(z2));  last layer writes d_out
    float* out = (l == NLAYER - 1) ? (float*)d_out : h;
    residual_bn_kernel<<<(int)((nd + 255) / 256), 256, 0, stream>>>(
        z2, scale, shift, h, out, (l < NLAYER - 1) ? 1 : 0, nd);
  }
}